// RBF_Conv2d_19275813224499
// MI455X (gfx1250) — compile-verified
//
#include <hip/hip_runtime.h>

typedef float v2f __attribute__((ext_vector_type(2)));
typedef float v8f __attribute__((ext_vector_type(8)));

#define KH 5
#define KW 5
#define TAPS 25
#define NB 16      // batch
#define OC 64      // output channels
#define HI 256
#define WI 256
#define HO 252
#define WO 252
#define NG 4       // channel groups of 16 (OC/16)
#define NK 7       // K steps of 4 (25 padded to 28)
#define TW 2       // column tiles of 16 per wave (32 pixels per workgroup)
#define NRED (NB*OC)  // 1024 values per pixel for std

__global__ __launch_bounds__(512) void rbf_conv2d_fused(
    const float* __restrict__ x,   // (16,1,256,256)
    const float* __restrict__ wgt, // (64,1,5,5)
    float* __restrict__ out)       // (16,64,252,252)
{
    __shared__ float s_wn[OC];
    __shared__ float s_r1[NB][TW * 16];
    __shared__ float s_r2[NB][TW * 16];

    const int tid  = threadIdx.x;
    const int lane = tid & 31;
    const int wave = tid >> 5;      // wave == batch index (16 waves)
    const int col  = lane & 15;     // pixel column within 16-wide tile / M index
    const int hf   = lane >> 4;     // wave half (0/1) -> K offset 2*hf, row offset 8*hf
    const int row  = blockIdx.y;    // output row
    const int w0   = blockIdx.x * (TW * 16);

    // ---- filter norms wn[o] into LDS (one thread per channel) ----
    if (tid < OC) {
        float s = 0.f;
        #pragma unroll
        for (int k = 0; k < TAPS; ++k) { float v = wgt[tid * TAPS + k]; s += v * v; }
        s_wn[tid] = s;
    }
    __syncthreads();

    // ---- A fragments: weights. A is 16(M=channel)x4(K); lane holds M=col,
    //      K = k0 + v + 2*hf for vgpr v. Zero-pad K>=25. ----
    v2f afrag[NG][NK];
    #pragma unroll
    for (int g = 0; g < NG; ++g) {
        #pragma unroll
        for (int s = 0; s < NK; ++s) {
            const int k0 = 4 * s + 2 * hf;
            const int c  = g * 16 + col;
            v2f a;
            a.x = (k0     < TAPS) ? wgt[c * TAPS + k0    ] : 0.f;
            a.y = (k0 + 1 < TAPS) ? wgt[c * TAPS + k0 + 1] : 0.f;
            afrag[g][s] = a;
        }
    }

    // ---- B fragments: patches for this wave's batch. B is 4(K)x16(N=pixel col);
    //      lane holds N=col, K = k0 + v + 2*hf. Accumulate ||patch||^2 on the fly. ----
    const float* xb = x + (size_t)wave * (HI * WI);
    v2f  bfrag[TW][NK];
    float psum[TW];
    #pragma unroll
    for (int t = 0; t < TW; ++t) {
        int wc = w0 + t * 16 + col;
        if (wc > WO - 1) wc = WO - 1;   // clamp masked edge columns (avoid OOB reads)
        psum[t] = 0.f;
        #pragma unroll
        for (int s = 0; s < NK; ++s) {
            const int k0 = 4 * s + 2 * hf;
            float b0 = 0.f, b1 = 0.f;
            if (k0     < TAPS) b0 = xb[(row + k0 / KW) * WI + wc + k0 % KW];
            if (k0 + 1 < TAPS) { const int k = k0 + 1; b1 = xb[(row + k / KW) * WI + wc + k % KW]; }
            v2f b; b.x = b0; b.y = b1;
            bfrag[t][s] = b;
            psum[t] += b0 * b0 + b1 * b1;
        }
    }
    float pn[TW];
    #pragma unroll
    for (int t = 0; t < TW; ++t)
        pn[t] = psum[t] + __shfl_xor(psum[t], 16, 32);   // fold the two K-halves

    // ---- GEMM: cross[o][w] for this batch, fp32 WMMA, K=28 (3 zero-padded) ----
    v8f acc[NG][TW];
    #pragma unroll
    for (int g = 0; g < NG; ++g) {
        #pragma unroll
        for (int t = 0; t < TW; ++t) {
            v8f c = {};
            #pragma unroll
            for (int s = 0; s < NK; ++s)
                c = __builtin_amdgcn_wmma_f32_16x16x4_f32(
                        false, afrag[g][s], false, bfrag[t][s],
                        (short)0, c, false, false);
            acc[g][t] = c;
        }
    }

    // ---- dist in-place, per-column partial sums for std ----
    float s1[TW] = {0.f, 0.f}, s2[TW] = {0.f, 0.f};
    #pragma unroll
    for (int g = 0; g < NG; ++g) {
        #pragma unroll
        for (int t = 0; t < TW; ++t) {
            #pragma unroll
            for (int v = 0; v < 8; ++v) {
                const float wn = s_wn[g * 16 + v + 8 * hf];  // channel for C row v
                float d2 = wn + pn[t] - 2.f * acc[g][t][v];
                d2 = fmaxf(d2, 1e-12f);
                const float d = __builtin_sqrtf(d2);
                acc[g][t][v] = d;                 // keep dist for the output pass
                s1[t] += d;
                s2[t] += d * d;
            }
        }
    }
    #pragma unroll
    for (int t = 0; t < TW; ++t) {
        s1[t] += __shfl_xor(s1[t], 16, 32);       // add partner half's 32 channels
        s2[t] += __shfl_xor(s2[t], 16, 32);
    }
    if (hf == 0) {
        #pragma unroll
        for (int t = 0; t < TW; ++t) {
            s_r1[wave][t * 16 + col] = s1[t];
            s_r2[wave][t * 16 + col] = s2[t];
        }
    }
    __syncthreads();

    // ---- per-column std over all 1024 (batch, channel) values; fixed-order
    //      (deterministic) sum over the 16 waves, computed redundantly per lane ----
    float n05_invvar[TW];
    #pragma unroll
    for (int t = 0; t < TW; ++t) {
        float S1 = 0.f, S2 = 0.f;
        #pragma unroll
        for (int b = 0; b < NB; ++b) {
            S1 += s_r1[b][t * 16 + col];
            S2 += s_r2[b][t * 16 + col];
        }
        const float mean = S1 * (1.f / NRED);
        const float var  = (S2 - (float)NRED * mean * mean) * (1.f / (NRED - 1));
        n05_invvar[t] = -0.5f / var;
    }

    // ---- stores: per (g,t,v,half) a 16-float contiguous 64B segment ----
    #pragma unroll
    for (int g = 0; g < NG; ++g) {
        #pragma unroll
        for (int t = 0; t < TW; ++t) {
            const int wc = w0 + t * 16 + col;
            if (wc < WO) {
                #pragma unroll
                for (int v = 0; v < 8; ++v) {
                    const int ch = g * 16 + v + 8 * hf;
                    const float d = acc[g][t][v];
                    out[((wave * OC + ch) * HO + row) * WO + wc] =
                        __expf(n05_invvar[t] * d * d);
                }
            }
        }
    }
}

extern "C" void kernel_launch(void* const* d_in, const int* in_sizes, int n_in,
                              void* d_out, int out_size, void* d_ws, size_t ws_size,
                              hipStream_t stream) {
    const float* x   = (const float*)d_in[0];   // (16,1,256,256) f32
    const float* wgt = (const float*)d_in[1];   // (64,1,5,5) f32
    float* out = (float*)d_out;                 // (16,64,252,252) f32
    dim3 grid((WO + TW * 16 - 1) / (TW * 16), HO);  // 8 x 252 workgroups
    rbf_conv2d_fused<<<grid, 512, 0, stream>>>(x, wgt, out);
}